// simpedge_23029614641734
// MI455X (gfx1250) — compile-verified
//
#include <hip/hip_runtime.h>

#define NB   512   // batches
#define NPG  512   // points per batch
#define KNN  8
#define FD   4
#define HD   15
#define SLOPE 0.01f

typedef __attribute__((ext_vector_type(2))) float v2f;
typedef __attribute__((ext_vector_type(8))) float v8f;

__device__ __forceinline__ float leaky(float v) {
    return (v >= 0.0f) ? v : SLOPE * v;
}

// ---------------------------------------------------------------------------
// Kernel 1: per-batch KNN (K=8, first 3 coords, self excluded).
// One block per batch; positions staged in LDS; per-thread sorted top-8.
// ---------------------------------------------------------------------------
__global__ __launch_bounds__(256) void knn_kernel(const float* __restrict__ x,
                                                  int* __restrict__ nbr)
{
    __shared__ float px[NPG], py[NPG], pz[NPG];
    const int b = blockIdx.x;
    const int t = threadIdx.x;
    const float* xb = x + (size_t)b * NPG * FD;

    for (int p = t; p < NPG; p += 256) {
        px[p] = xb[p * 4 + 0];
        py[p] = xb[p * 4 + 1];
        pz[p] = xb[p * 4 + 2];
    }
    __syncthreads();

    for (int p = t; p < NPG; p += 256) {
        const float qx = px[p], qy = py[p], qz = pz[p];
        float dist[KNN];
        int   ind[KNN];
#pragma unroll
        for (int k = 0; k < KNN; ++k) { dist[k] = 3.0e38f; ind[k] = 0; }

        for (int j = 0; j < NPG; ++j) {
            const float dx = px[j] - qx;
            const float dy = py[j] - qy;
            const float dz = pz[j] - qz;
            const float d  = dx * dx + dy * dy + dz * dz;
            if (j != p && d < dist[KNN - 1]) {
                dist[KNN - 1] = d; ind[KNN - 1] = j;
#pragma unroll
                for (int s = KNN - 1; s > 0; --s) {
                    if (dist[s] < dist[s - 1]) {
                        float td = dist[s]; dist[s] = dist[s - 1]; dist[s - 1] = td;
                        int   ti = ind[s];  ind[s]  = ind[s - 1];  ind[s - 1]  = ti;
                    }
                }
            }
        }
        int* o = nbr + ((size_t)b * NPG + p) * KNN;
#pragma unroll
        for (int k = 0; k < KNN; ++k) o[k] = ind[k];
    }
}

// ---------------------------------------------------------------------------
// Kernel 2: edge MLP via V_WMMA_F32_16X16X4_F32 + per-batch reduction.
// One block (256 thr = 8 waves) per batch; each wave: 2 points / 16 edges per
// WMMA tile, 32 tiles. fp32 WMMA keeps reference-level accuracy (K=8,16 small).
// ---------------------------------------------------------------------------
__global__ __launch_bounds__(256) void edge_mlp_kernel(
    const float* __restrict__ x,
    const float* __restrict__ W1, const float* __restrict__ b1,
    const float* __restrict__ W2, const float* __restrict__ b2,
    const float* __restrict__ Wn1, const float* __restrict__ bn1,
    const float* __restrict__ Wn2, const float* __restrict__ bn2,
    const int* __restrict__ nbr, float* __restrict__ out)
{
    __shared__ float4 xs[NPG];          // 8 KB: per-batch features
    __shared__ float  w1p[16][16];      // W1 zero-padded, K-major
    __shared__ float  w2p[16][16];      // W2 zero-padded, K-major
    __shared__ float  h1s[8][16][17];   // per-wave layer-1 tile (padded stride)
    __shared__ float  redmax[8], redmin[8], redsum[8];

    const int  b  = blockIdx.x;
    const int  t  = threadIdx.x;
    const int  w  = t >> 5;             // wave id 0..7
    const int  l  = t & 31;             // lane
    const int  n  = l & 15;             // tile column / tile row index
    const bool hi = (l >= 16);
    const int  c0 = hi ? 2 : 0;         // K-pair selected by lane half

    // stage features + padded weights
    const float4* xg = (const float4*)x + (size_t)b * NPG;
    for (int p = t; p < NPG; p += 256) xs[p] = xg[p];
    {
        const int r = t >> 4, c = t & 15;
        w1p[r][c] = (r < 2 * FD && c < HD) ? W1[r * HD + c] : 0.0f;
        w2p[r][c] = (r < HD     && c < HD) ? W2[r * HD + c] : 0.0f;
    }
    __syncthreads();

    // B operands: B(4x16) layout — lanes 0..15 hold K=0(V0),1(V1); lanes 16..31 K=2,3
    const v2f B10 = { w1p[c0 + 0][n],  w1p[c0 + 1][n]  };
    const v2f B11 = { w1p[c0 + 4][n],  w1p[c0 + 5][n]  };
    const v2f B20 = { w2p[c0 + 0][n],  w2p[c0 + 1][n]  };
    const v2f B21 = { w2p[c0 + 4][n],  w2p[c0 + 5][n]  };
    const v2f B22 = { w2p[c0 + 8][n],  w2p[c0 + 9][n]  };
    const v2f B23 = { w2p[c0 + 12][n], w2p[c0 + 13][n] };
    const float b1v  = (n < HD) ? b1[n]  : 0.0f;
    const float b2v  = (n < HD) ? b2[n]  : 0.0f;
    const float wn1v = (n < HD) ? Wn1[n] : 0.0f;
    const float bn1v = bn1[0];

    float vmax = -3.0e38f, vmin = 3.0e38f, vsum = 0.0f;

    // A-row r = edge index within tile: rows 0..7 -> point pbase (+its 8 nbrs),
    // rows 8..15 -> point pbase+1.
    const int r     = n;
    const int kslot = r & 7;
    const int rsel  = r >> 3;

    for (int it = 0; it < 32; ++it) {
        const int pbase = w * 64 + it * 2;
        const int p = pbase + rsel;
        const int j = nbr[((size_t)b * NPG + p) * KNN + kslot];
        const float4 xi4 = xs[p];
        const float4 xj4 = xs[j];
        const float xi0 = hi ? xi4.z : xi4.x;
        const float xi1 = hi ? xi4.w : xi4.y;
        const float xd0 = (hi ? xj4.z : xj4.x) - xi0;
        const float xd1 = (hi ? xj4.w : xj4.y) - xi1;

        // ---- layer 1: m(16x8) @ W1(8x16pad), K split 2x4 ----
        v2f A0 = { xi0, xi1 };          // m cols 0..3 = xi
        v2f A1 = { xd0, xd1 };          // m cols 4..7 = xj - xi
        v8f acc = {0.f,0.f,0.f,0.f,0.f,0.f,0.f,0.f};
        acc = __builtin_amdgcn_wmma_f32_16x16x4_f32(false, A0, false, B10, (short)0, acc, false, false);
        acc = __builtin_amdgcn_wmma_f32_16x16x4_f32(false, A1, false, B11, (short)0, acc, false, false);
#pragma unroll
        for (int v = 0; v < 8; ++v) {
            const float h = leaky(acc[v] + b1v);
            h1s[w][v + (hi ? 8 : 0)][n] = h;      // C row = v (+8 for hi half)
        }

        // ---- layer 2: h1(16x16) @ W2(16x16pad), K split 4x4 ----
        v8f acc2 = {0.f,0.f,0.f,0.f,0.f,0.f,0.f,0.f};
        v2f A;
        A[0] = h1s[w][r][c0 + 0];  A[1] = h1s[w][r][c0 + 1];
        acc2 = __builtin_amdgcn_wmma_f32_16x16x4_f32(false, A, false, B20, (short)0, acc2, false, false);
        A[0] = h1s[w][r][c0 + 4];  A[1] = h1s[w][r][c0 + 5];
        acc2 = __builtin_amdgcn_wmma_f32_16x16x4_f32(false, A, false, B21, (short)0, acc2, false, false);
        A[0] = h1s[w][r][c0 + 8];  A[1] = h1s[w][r][c0 + 9];
        acc2 = __builtin_amdgcn_wmma_f32_16x16x4_f32(false, A, false, B22, (short)0, acc2, false, false);
        A[0] = h1s[w][r][c0 + 12]; A[1] = h1s[w][r][c0 + 13];
        acc2 = __builtin_amdgcn_wmma_f32_16x16x4_f32(false, A, false, B23, (short)0, acc2, false, false);

        // per-lane sum of 8 C rows == sum over this point's K=8 neighbors
        float s = 0.0f;
#pragma unroll
        for (int v = 0; v < 8; ++v) s += leaky(acc2[v] + b2v);

        // y[p] = leaky(xnew . Wn1 + bn1): reduce over 16 channels per half-wave
        float ty = s * wn1v;
#pragma unroll
        for (int off = 1; off < 16; off <<= 1) ty += __shfl_xor(ty, off, 32);
        const float y = leaky(ty + bn1v);
        // lanes 0..15 hold y(pbase), lanes 16..31 hold y(pbase+1)
        vmax = fmaxf(vmax, y);
        vmin = fminf(vmin, y);
        vsum += y;
    }

    // combine half-wave streams, then waves
    vmax = fmaxf(vmax, __shfl_xor(vmax, 16, 32));
    vmin = fminf(vmin, __shfl_xor(vmin, 16, 32));
    vsum = vsum + __shfl_xor(vsum, 16, 32);
    if (l == 0) { redmax[w] = vmax; redmin[w] = vmin; redsum[w] = vsum; }
    __syncthreads();

    if (t == 0) {
        float gmax = -3.0e38f, gmin = 3.0e38f, gsum = 0.0f;
#pragma unroll
        for (int i = 0; i < 8; ++i) {
            gmax = fmaxf(gmax, redmax[i]);
            gmin = fminf(gmin, redmin[i]);
            gsum += redsum[i];
        }
        const float gmean = gsum * (1.0f / (float)NPG);
        out[b] = gmax * Wn2[0] + gmin * Wn2[1] + gsum * Wn2[2] + gmean * Wn2[3] + bn2[0];
    }
}

// ---------------------------------------------------------------------------
extern "C" void kernel_launch(void* const* d_in, const int* in_sizes, int n_in,
                              void* d_out, int out_size, void* d_ws, size_t ws_size,
                              hipStream_t stream)
{
    const float* x   = (const float*)d_in[0];
    const float* W1  = (const float*)d_in[1];
    const float* b1  = (const float*)d_in[2];
    const float* W2  = (const float*)d_in[3];
    const float* b2  = (const float*)d_in[4];
    const float* Wn1 = (const float*)d_in[5];
    const float* bn1 = (const float*)d_in[6];
    const float* Wn2 = (const float*)d_in[7];
    const float* bn2 = (const float*)d_in[8];
    // d_in[9] = batch indices (implicit in layout; unused)

    int* nbr = (int*)d_ws;              // NB*NPG*KNN ints = 8 MB scratch
    float* out = (float*)d_out;

    knn_kernel<<<NB, 256, 0, stream>>>(x, nbr);
    edge_mlp_kernel<<<NB, 256, 0, stream>>>(x, W1, b1, W2, b2, Wn1, bn1,
                                            Wn2, bn2, nbr, out);
}